// GCN_79542794322476
// MI455X (gfx1250) — compile-verified
//
#include <hip/hip_runtime.h>
#include <hip/hip_bf16.h>

typedef __attribute__((ext_vector_type(2))) float v2f;
typedef __attribute__((ext_vector_type(8))) float v8f;

#define IN_DIM 128
#define HID_DIM 64
#define NCLS 8

// ---------------------------------------------------------------- utilities

__global__ void gcn_fill_f32(float* __restrict__ p, long n, float v) {
    long i = (long)blockIdx.x * blockDim.x + threadIdx.x;
    if (i < n) p[i] = v;
}

// degree accumulation: one thread per edge, two f32 atomics
__global__ void gcn_degree(const int* __restrict__ src, const int* __restrict__ dst,
                           float* __restrict__ out_deg, float* __restrict__ in_deg, int E) {
    int e = blockIdx.x * blockDim.x + threadIdx.x;
    if (e < E) {
        atomicAdd(&out_deg[src[e]], 1.0f);
        atomicAdd(&in_deg[dst[e]], 1.0f);
    }
}

// in-place deg -> rsqrt(max(deg,1))
__global__ void gcn_norm(float* __restrict__ out_deg, float* __restrict__ in_deg, int N) {
    int i = blockIdx.x * blockDim.x + threadIdx.x;
    if (i < N) {
        out_deg[i] = rsqrtf(fmaxf(out_deg[i], 1.0f));
        in_deg[i]  = rsqrtf(fmaxf(in_deg[i],  1.0f));
    }
}

// ---------------------------------------------------------------- GEMM 1
// H1[16-row tile] = (X * out_norm[:,None]) @ W1   (128 -> 64), f32 WMMA.
// W1 is staged in LDS in K-pair-interleaved layout so each B fragment is a
// single 8-byte ds_load_b64 directly into the WMMA source register pair.
__global__ void gcn_gemm1_wmma(const float* __restrict__ X, const float* __restrict__ W1,
                               const float* __restrict__ onorm, float* __restrict__ H1,
                               int N) {
    // w[kp*HID + col] = { W1[2kp][col], W1[2kp+1][col] }   (32 KB)
    __shared__ float2 w[(IN_DIM / 2) * HID_DIM];
    for (int i = threadIdx.x; i < (IN_DIM / 2) * HID_DIM; i += blockDim.x) {
        int kp = i >> 6, col = i & 63;
        w[i] = make_float2(W1[(2 * kp) * HID_DIM + col],
                           (float)W1[(2 * kp + 1) * HID_DIM + col]);
    }
    __syncthreads();

    const int wave  = threadIdx.x >> 5;
    const int lane  = threadIdx.x & 31;
    const int mTile = blockIdx.x * (blockDim.x >> 5) + wave;
    if (mTile * 16 >= N) return;                    // wave-uniform: EXEC stays full

    const int mBase = mTile * 16;
    const int m     = mBase + (lane & 15);
    const int khalf = (lane >> 4) << 1;             // lanes 0-15 -> K {0,1}; 16-31 -> K {2,3}
    const int c     = lane & 15;                    // B/N column within tile
    const float scale = onorm[m];
    const float* xrow = X + (long)m * IN_DIM;

    v8f acc0 = {}, acc1 = {}, acc2 = {}, acc3 = {};
    #pragma unroll 4
    for (int k = 0; k < IN_DIM; k += 4) {
        v2f a = *(const v2f*)(xrow + k + khalf);    // global_load_b64
        a *= scale;                                 // v_pk_mul_f32 (folds D^-1/2)
        const float2* wr = &w[((k + khalf) >> 1) * HID_DIM];
        v2f b0 = *(const v2f*)&wr[c];               // ds_load_b64 each
        v2f b1 = *(const v2f*)&wr[16 + c];
        v2f b2 = *(const v2f*)&wr[32 + c];
        v2f b3 = *(const v2f*)&wr[48 + c];
        acc0 = __builtin_amdgcn_wmma_f32_16x16x4_f32(false, a, false, b0, (short)0, acc0, false, false);
        acc1 = __builtin_amdgcn_wmma_f32_16x16x4_f32(false, a, false, b1, (short)0, acc1, false, false);
        acc2 = __builtin_amdgcn_wmma_f32_16x16x4_f32(false, a, false, b2, (short)0, acc2, false, false);
        acc3 = __builtin_amdgcn_wmma_f32_16x16x4_f32(false, a, false, b3, (short)0, acc3, false, false);
    }

    // C/D layout: VGPR i -> (M = mBase + 8*(lane>=16) + i, Ncol = lane%16)
    const int rowoff = (lane >> 4) << 3;
    #pragma unroll
    for (int i = 0; i < 8; ++i) {
        float* hr = H1 + (long)(mBase + rowoff + i) * HID_DIM;
        hr[c]      = acc0[i];
        hr[16 + c] = acc1[i];
        hr[32 + c] = acc2[i];
        hr[48 + c] = acc3[i];
    }
}

// ---------------------------------------------------------------- edge aggregation
// 64 threads per edge: wave-uniform src/dst, 128B coalesced gather + atomics
__global__ void gcn_agg64(const int* __restrict__ src, const int* __restrict__ dst,
                          const float* __restrict__ H, float* __restrict__ AGG, long total) {
    long gid = (long)blockIdx.x * blockDim.x + threadIdx.x;
    if (gid >= total) return;
    int e = (int)(gid >> 6);
    int ch = (int)(gid & 63);
    int s = src[e], d = dst[e];
    atomicAdd(&AGG[(long)d * HID_DIM + ch], H[(long)s * HID_DIM + ch]);
}

__global__ void gcn_agg8(const int* __restrict__ src, const int* __restrict__ dst,
                         const float* __restrict__ H, float* __restrict__ AGG, long total) {
    long gid = (long)blockIdx.x * blockDim.x + threadIdx.x;
    if (gid >= total) return;
    int e = (int)(gid >> 3);
    int ch = (int)(gid & 7);
    int s = src[e], d = dst[e];
    atomicAdd(&AGG[(long)d * NCLS + ch], H[(long)s * NCLS + ch]);
}

// H2 = relu(AGG1 * in_norm + b1)
__global__ void gcn_bias_relu(const float* __restrict__ AGG, const float* __restrict__ inorm,
                              const float* __restrict__ b1, float* __restrict__ H2, long n) {
    long i = (long)blockIdx.x * blockDim.x + threadIdx.x;
    if (i < n) {
        int node = (int)(i >> 6);
        int ch = (int)(i & 63);
        float v = fmaf(AGG[i], inorm[node], b1[ch]);
        H2[i] = v > 0.0f ? v : 0.0f;
    }
}

// ---------------------------------------------------------------- GEMM 2
// H3 = H2 @ W2  (64 -> 8), classes padded to 16 zero cols, K-pair LDS layout
__global__ void gcn_gemm2_wmma(const float* __restrict__ H2, const float* __restrict__ W2,
                               float* __restrict__ H3, int N) {
    // w[kp*16 + col] = { W2pad[2kp][col], W2pad[2kp+1][col] }   (4 KB)
    __shared__ float2 w[(HID_DIM / 2) * 16];
    for (int i = threadIdx.x; i < (HID_DIM / 2) * 16; i += blockDim.x) {
        int kp = i >> 4, cc = i & 15;
        float lo = (cc < NCLS) ? W2[(2 * kp) * NCLS + cc] : 0.0f;
        float hi = (cc < NCLS) ? W2[(2 * kp + 1) * NCLS + cc] : 0.0f;
        w[i] = make_float2(lo, hi);
    }
    __syncthreads();

    const int wave  = threadIdx.x >> 5;
    const int lane  = threadIdx.x & 31;
    const int mTile = blockIdx.x * (blockDim.x >> 5) + wave;
    if (mTile * 16 >= N) return;

    const int mBase = mTile * 16;
    const int m     = mBase + (lane & 15);
    const int khalf = (lane >> 4) << 1;
    const int c     = lane & 15;
    const float* hrow = H2 + (long)m * HID_DIM;

    v8f acc = {};
    #pragma unroll 4
    for (int k = 0; k < HID_DIM; k += 4) {
        v2f a = *(const v2f*)(hrow + k + khalf);
        v2f b = *(const v2f*)&w[((k + khalf) >> 1) * 16 + c];
        acc = __builtin_amdgcn_wmma_f32_16x16x4_f32(false, a, false, b, (short)0, acc, false, false);
    }

    const int rowoff = (lane >> 4) << 3;
    if (c < NCLS) {                                  // predicated stores after all WMMAs
        #pragma unroll
        for (int i = 0; i < 8; ++i)
            H3[(long)(mBase + rowoff + i) * NCLS + c] = acc[i];
    }
}

// out = softmax(AGG2 * in_norm + b2); one node per thread, 32B contiguous per thread
__global__ void gcn_final_softmax(const float* __restrict__ AGG2, const float* __restrict__ inorm,
                                  const float* __restrict__ b2, float* __restrict__ out, int N) {
    int i = blockIdx.x * blockDim.x + threadIdx.x;
    if (i >= N) return;
    float nn = inorm[i];
    float v[NCLS];
    float mx = -3.402823466e+38f;
    #pragma unroll
    for (int ch = 0; ch < NCLS; ++ch) {
        v[ch] = fmaf(AGG2[(long)i * NCLS + ch], nn, b2[ch]);
        mx = fmaxf(mx, v[ch]);
    }
    float s = 0.0f;
    #pragma unroll
    for (int ch = 0; ch < NCLS; ++ch) { v[ch] = __expf(v[ch] - mx); s += v[ch]; }
    float inv = 1.0f / s;
    #pragma unroll
    for (int ch = 0; ch < NCLS; ++ch) out[(long)i * NCLS + ch] = v[ch] * inv;
}

// ---------------------------------------------------------------- launch

extern "C" void kernel_launch(void* const* d_in, const int* in_sizes, int n_in,
                              void* d_out, int out_size, void* d_ws, size_t ws_size,
                              hipStream_t stream) {
    const float* x   = (const float*)d_in[0];
    const int*   src = (const int*)  d_in[1];
    const int*   dst = (const int*)  d_in[2];
    const float* W1  = (const float*)d_in[3];
    const float* b1  = (const float*)d_in[4];
    const float* W2  = (const float*)d_in[5];
    const float* b2  = (const float*)d_in[6];
    float* out = (float*)d_out;

    const int N = in_sizes[0] / IN_DIM;
    const int E = in_sizes[1];

    // workspace layout (floats): [onorm N][inorm N][bufA 64N][bufB 64N]
    float* ws    = (float*)d_ws;
    float* onorm = ws;
    float* inorm = ws + (long)N;
    float* bufA  = ws + 2L * N;               // H1, then H2
    float* bufB  = bufA + 64L * N;            // AGG1, later reused:
    float* agg2  = bufB;                      //   AGG2: 8N
    float* h3    = bufB + 8L * N;             //   H3:   8N

    const int T = 256;
    auto blocks = [](long n, int t) { return (unsigned)((n + t - 1) / t); };

    // degrees -> norms
    gcn_fill_f32<<<blocks(2L * N, T), T, 0, stream>>>(onorm, 2L * N, 0.0f);
    gcn_degree<<<blocks(E, T), T, 0, stream>>>(src, dst, onorm, inorm, E);
    gcn_norm<<<blocks(N, T), T, 0, stream>>>(onorm, inorm, N);

    // layer 1: project (WMMA), aggregate, normalize+bias+relu
    const int mTiles = (N + 15) / 16;
    gcn_gemm1_wmma<<<(mTiles + 7) / 8, T, 0, stream>>>(x, W1, onorm, bufA, N);
    gcn_fill_f32<<<blocks(64L * N, T), T, 0, stream>>>(bufB, 64L * N, 0.0f);
    gcn_agg64<<<blocks((long)E * 64, T), T, 0, stream>>>(src, dst, bufA, bufB, (long)E * 64);
    gcn_bias_relu<<<blocks(64L * N, T), T, 0, stream>>>(bufB, inorm, b1, bufA, 64L * N);

    // layer 2: project (WMMA), aggregate, normalize+bias+softmax
    gcn_gemm2_wmma<<<(mTiles + 7) / 8, T, 0, stream>>>(bufA, W2, h3, N);
    gcn_fill_f32<<<blocks(8L * N, T), T, 0, stream>>>(agg2, 8L * N, 0.0f);
    gcn_agg8<<<blocks((long)E * 8, T), T, 0, stream>>>(src, dst, h3, agg2, (long)E * 8);
    gcn_final_softmax<<<blocks(N, T), T, 0, stream>>>(agg2, inorm, b2, out, N);
}